// Gru_58025008169266
// MI455X (gfx1250) — compile-verified
//
#include <hip/hip_runtime.h>
#include <math.h>

typedef __attribute__((ext_vector_type(2))) float v2f;
typedef __attribute__((ext_vector_type(8))) float v8f;
typedef int v4i __attribute__((vector_size(16)));   // matches builtin signature

#define B_ 64
#define T_ 1024
#define I_ 256
#define H_ 512
#define K_ 768            // I + H
#define BTH_ ((size_t)B_ * T_ * H_)

// ---------------------------------------------------------------------------
// CDNA5 async global->LDS copy path (ASYNCcnt-tracked), with VGPR fallback.
// ---------------------------------------------------------------------------
#if defined(__has_builtin)
#if __has_builtin(__builtin_amdgcn_global_load_async_to_lds_b128) && \
    __has_builtin(__builtin_amdgcn_s_wait_asynccnt)
#define HAVE_ASYNC_LDS 1
#endif
#endif

__device__ __forceinline__ void cp_lds_16B(float* lds_dst, const float* gsrc) {
#if defined(HAVE_ASYNC_LDS)
    __builtin_amdgcn_global_load_async_to_lds_b128(
        (__attribute__((address_space(1))) v4i*)(gsrc),
        (__attribute__((address_space(3))) v4i*)(lds_dst), 0, 0);
#else
    *(float4*)lds_dst = *(const float4*)gsrc;
#endif
}

__device__ __forceinline__ void stage_fence() {
#if defined(HAVE_ASYNC_LDS)
    __builtin_amdgcn_s_wait_asynccnt(0);
#endif
    __syncthreads();
}

__device__ __forceinline__ float fast_sigmoid(float x) {
    return __builtin_amdgcn_rcpf(1.0f + __expf(-x));
}

__device__ __forceinline__ float fast_tanh(float x) {
#if defined(__has_builtin) && __has_builtin(__builtin_amdgcn_tanhf)
    return __builtin_amdgcn_tanhf(x);
#else
    const float e = __expf(-2.0f * x);
    return (1.0f - e) * __builtin_amdgcn_rcpf(1.0f + e);
#endif
}

// One 16x16 fp32 WMMA K-step: A row-strip in LDS (row stride KA), B in global.
__device__ __forceinline__ v8f wmma_k4(v8f acc, const float* sA_row, int k0,
                                       const float* W, int n, int hi) {
    v2f a, b;
    const float* ap = sA_row + k0 + 2 * hi;       // lanes0-15: K=k0,k0+1; 16-31: +2,+3
    a.x = ap[0]; a.y = ap[1];
    b.x = W[(size_t)(k0 + hi) * H_ + n];          // v0 rows K=k0(+hi)
    b.y = W[(size_t)(k0 + 2 + hi) * H_ + n];      // v1 rows K=k0+2(+hi)
    return __builtin_amdgcn_wmma_f32_16x16x4_f32(
        false, a, false, b, (short)0, acc, false, false);
}

// ---------------------------------------------------------------------------
// Fully parallel input projection: xp[g] = xs @ Wg[0:I,:] + bg  (g = z,r,h)
// grid = (n_strip=4, m_tile=4096, gate=3), 256 threads = 8 waves,
// one 16x16 tile per wave, K = 256 (64 WMMAs).
// ---------------------------------------------------------------------------
__global__ __launch_bounds__(256) void gru_xproj(
    const float* __restrict__ xs,
    const float* __restrict__ Wz, const float* __restrict__ bz,
    const float* __restrict__ Wr, const float* __restrict__ br,
    const float* __restrict__ Wh, const float* __restrict__ bh,
    float* __restrict__ xp)                       // [3][B*T][H]
{
    __shared__ float sA[16 * I_];                 // 16 KB
    const int n_strip = blockIdx.x;
    const int m_tile  = blockIdx.y;
    const int gate    = blockIdx.z;
    const size_t m0   = (size_t)m_tile * 16;

    const float* __restrict__ W    = gate == 0 ? Wz : (gate == 1 ? Wr : Wh);
    const float* __restrict__ bias = gate == 0 ? bz : (gate == 1 ? br : bh);

    // Stage 16x256 A-strip (pure copy -> async-to-LDS path).
    for (int i = threadIdx.x; i < 16 * (I_ / 4); i += 256) {
        const int m  = i >> 6;                    // /64
        const int kv = i & 63;
        cp_lds_16B(&sA[m * I_ + kv * 4], &xs[(m0 + m) * I_ + kv * 4]);
    }
    stage_fence();

    const int wave = threadIdx.x >> 5;
    const int lane = threadIdx.x & 31;
    const int hi   = lane >> 4;
    const int lo   = lane & 15;
    const int n    = n_strip * 128 + wave * 16 + lo;

    v8f acc = {};
    const float* sArow = &sA[lo * I_];
    for (int k0 = 0; k0 < I_; k0 += 4)
        acc = wmma_k4(acc, sArow, k0, W, n, hi);

    const float bn = bias[n];
    float* __restrict__ og = xp + (size_t)gate * BTH_;
#pragma unroll
    for (int v = 0; v < 8; ++v) {
        const size_t m = m0 + v + 8 * hi;
        og[m * H_ + n] = acc[v] + bn;
    }
}

// ---------------------------------------------------------------------------
// Per-step phase 1: z,r gates.  PRE: A = h (K=512), x-proj added in epilogue.
// !PRE: A = [x_t | h] (K=768), bias added in epilogue.
// grid = 32 (gate[2] x m_tile[4] x n_strip[4]), 256 threads = 8 waves.
// NOTE (PRE): Wz/Wr passed pre-offset by I_*H_ rows.
// ---------------------------------------------------------------------------
template <bool PRE>
__global__ __launch_bounds__(256) void gru_gates(
    const float* __restrict__ xs,
    const float* __restrict__ Wz, const float* __restrict__ bz,
    const float* __restrict__ Wr, const float* __restrict__ br,
    const float* __restrict__ hs_out,             // d_out hs region (h_{t-1})
    const float* __restrict__ xp,                 // [3][B*T][H] (PRE only)
    float* __restrict__ zr_ws,                    // [2][B_][H_]
    int t)
{
    constexpr int KA = PRE ? H_ : K_;
    constexpr int VR = KA / 4;                    // float4s per row
    __shared__ float sA[16 * KA];

    const int bid     = blockIdx.x;
    const int gate    = bid >> 4;
    const int m_tile  = (bid >> 2) & 3;
    const int n_strip = bid & 3;
    const int m0      = m_tile * 16;

    const float* __restrict__ W    = gate ? Wr : Wz;
    const float* __restrict__ bias = gate ? br : bz;

    for (int i = threadIdx.x; i < 16 * VR; i += 256) {
        const int m  = i / VR;
        const int kv = i - m * VR;
        const int b  = m0 + m;
        float* dst = &sA[m * KA + kv * 4];
        if (!PRE && kv < I_ / 4) {
            cp_lds_16B(dst, &xs[((size_t)b * T_ + t) * I_ + kv * 4]);
        } else {
            const int j = (PRE ? kv : kv - I_ / 4) * 4;
            if (t == 0) {
                float4 z; z.x = z.y = z.z = z.w = 0.0f;
                *(float4*)dst = z;
            } else {
                cp_lds_16B(dst, &hs_out[((size_t)b * T_ + (t - 1)) * H_ + j]);
            }
        }
    }
    stage_fence();

    const int wave = threadIdx.x >> 5;
    const int lane = threadIdx.x & 31;
    const int hi   = lane >> 4;
    const int lo   = lane & 15;
    const int n    = n_strip * 128 + wave * 16 + lo;

    v8f acc = {};
    const float* sArow = &sA[lo * KA];
    for (int k0 = 0; k0 < KA; k0 += 4)
        acc = wmma_k4(acc, sArow, k0, W, n, hi);

    const float bn = PRE ? 0.0f : bias[n];
    const float* __restrict__ xg = xp + (size_t)gate * BTH_;
    float* __restrict__ outg = zr_ws + (size_t)gate * B_ * H_;
#pragma unroll
    for (int v = 0; v < 8; ++v) {
        const int m = m0 + v + 8 * hi;
        const float pre = PRE ? xg[((size_t)m * T_ + t) * H_ + n] : bn;
        outg[(size_t)m * H_ + n] = fast_sigmoid(acc[v] + pre);
    }
}

// ---------------------------------------------------------------------------
// Per-step phase 2: h~ = tanh(xh_t + (r.h)@Wh_h); h' = (1-z)h + z h~
// PRE: A = r.h (K=512).  !PRE: A = [x_t | r.h] (K=768).
// grid = 16 (m_tile[4] x n_strip[4]).  NOTE (PRE): Wh pre-offset by I_*H_.
// ---------------------------------------------------------------------------
template <bool PRE>
__global__ __launch_bounds__(256) void gru_hidden(
    const float* __restrict__ xs,
    const float* __restrict__ Wh, const float* __restrict__ bh,
    const float* __restrict__ xp,                 // [3][B*T][H] (PRE only)
    const float* __restrict__ zr_ws,              // [2][B_][H_]
    float* __restrict__ out,                      // hs[B,T,H] then h_last[B,H]
    int t)
{
    constexpr int KA = PRE ? H_ : K_;
    constexpr int VR = KA / 4;
    __shared__ float sA[16 * KA];

    const int bid     = blockIdx.x;
    const int m_tile  = bid >> 2;
    const int n_strip = bid & 3;
    const int m0      = m_tile * 16;

    const float* __restrict__ r_ws = zr_ws + (size_t)B_ * H_;
    const float* __restrict__ hs   = out;

    for (int i = threadIdx.x; i < 16 * VR; i += 256) {
        const int m  = i / VR;
        const int kv = i - m * VR;
        const int b  = m0 + m;
        float* dst = &sA[m * KA + kv * 4];
        if (!PRE && kv < I_ / 4) {
            cp_lds_16B(dst, &xs[((size_t)b * T_ + t) * I_ + kv * 4]);
        } else {
            const int j = (PRE ? kv : kv - I_ / 4) * 4;
            float4 val; val.x = val.y = val.z = val.w = 0.0f;
            if (t != 0) {
                const float4 r4 = *(const float4*)&r_ws[(size_t)b * H_ + j];
                const float4 h4 =
                    *(const float4*)&hs[((size_t)b * T_ + (t - 1)) * H_ + j];
                val.x = r4.x * h4.x; val.y = r4.y * h4.y;
                val.z = r4.z * h4.z; val.w = r4.w * h4.w;
            }
            *(float4*)dst = val;
        }
    }
    stage_fence();

    const int wave = threadIdx.x >> 5;
    const int lane = threadIdx.x & 31;
    const int hi   = lane >> 4;
    const int lo   = lane & 15;
    const int n    = n_strip * 128 + wave * 16 + lo;

    v8f acc = {};
    const float* sArow = &sA[lo * KA];
    for (int k0 = 0; k0 < KA; k0 += 4)
        acc = wmma_k4(acc, sArow, k0, Wh, n, hi);

    const float bn = PRE ? 0.0f : bh[n];
    const float* __restrict__ xh = xp + 2 * BTH_;
#pragma unroll
    for (int v = 0; v < 8; ++v) {
        const int m = m0 + v + 8 * hi;
        const float pre  = PRE ? xh[((size_t)m * T_ + t) * H_ + n] : bn;
        const float htil = fast_tanh(acc[v] + pre);
        const float z    = zr_ws[(size_t)m * H_ + n];
        const float hp   = (t == 0) ? 0.0f
            : hs[((size_t)m * T_ + (t - 1)) * H_ + n];
        const float hn = (1.0f - z) * hp + z * htil;
        out[((size_t)m * T_ + t) * H_ + n] = hn;
        if (t == T_ - 1)
            out[BTH_ + (size_t)m * H_ + n] = hn;  // h_last
    }
}

// ---------------------------------------------------------------------------
extern "C" void kernel_launch(void* const* d_in, const int* in_sizes, int n_in,
                              void* d_out, int out_size, void* d_ws, size_t ws_size,
                              hipStream_t stream) {
    (void)in_sizes; (void)n_in; (void)out_size;
    const float* xs = (const float*)d_in[0];
    const float* Wz = (const float*)d_in[1];
    const float* bz = (const float*)d_in[2];
    const float* Wr = (const float*)d_in[3];
    const float* br = (const float*)d_in[4];
    const float* Wh = (const float*)d_in[5];
    const float* bh = (const float*)d_in[6];
    float* out = (float*)d_out;
    float* ws  = (float*)d_ws;

    const size_t need_pre = 3 * BTH_ * sizeof(float)
                          + (size_t)2 * B_ * H_ * sizeof(float);
    const bool pre = ws_size >= need_pre;

    if (pre) {
        float* xp = ws;                           // [3][B*T][H]
        float* zr = ws + 3 * BTH_;                // [2][B][H]
        const size_t off = (size_t)I_ * H_;       // skip x-rows of W
        gru_xproj<<<dim3(4, 4096, 3), 256, 0, stream>>>(
            xs, Wz, bz, Wr, br, Wh, bh, xp);
        for (int t = 0; t < T_; ++t) {
            gru_gates<true><<<32, 256, 0, stream>>>(
                xs, Wz + off, bz, Wr + off, br, out, xp, zr, t);
            gru_hidden<true><<<16, 256, 0, stream>>>(
                xs, Wh + off, bh, xp, zr, out, t);
        }
    } else {
        float* zr = ws;                           // [2][B][H] (256 KB)
        for (int t = 0; t < T_; ++t) {
            gru_gates<false><<<32, 256, 0, stream>>>(
                xs, Wz, bz, Wr, br, out, (const float*)ws, zr, t);
            gru_hidden<false><<<16, 256, 0, stream>>>(
                xs, Wh, bh, (const float*)ws, zr, out, t);
        }
    }
}